// PHMLinear_33938831573495
// MI455X (gfx1250) — compile-verified
//
#include <hip/hip_runtime.h>

// ---- problem constants (match reference) ----
#define LNUM   12
#define PDIM   4
#define IN_F   1024
#define OUT_F  1024
#define IN_P   256
#define OUT_P  256
#define NBLK   (2*LNUM*PDIM)     // 96
#define JDIM   (NBLK*PDIM)       // 384  (columns of T: j = n*4 + p_o)
#define ROWS   128
#define OUT_COLS (2*LNUM*OUT_F)  // 24576

typedef __attribute__((ext_vector_type(2))) float v2f;
typedef __attribute__((ext_vector_type(8))) float v8f;

// -------------------------------------------------------------------------
// Kernel 1: build Bhat[j][k] = rule[(n%8)][p_i][p_o] * Wl[n][i]
//   j = n*4 + p_o  in [0,384),  k = p_i*256 + i in [0,1024)
// Stored j-major (row = j, 1024 contiguous k) so GEMM-1 B-loads are b64.
// -------------------------------------------------------------------------
__global__ void phm_build_B(const float* __restrict__ Wl,
                            const float* __restrict__ rule,
                            float* __restrict__ Bhat) {
    int idx = blockIdx.x * blockDim.x + threadIdx.x;   // 0 .. 384*1024-1
    int j   = idx >> 10;
    int k   = idx & 1023;
    int p_i = k >> 8;
    int i   = k & 255;
    int n   = j >> 2;
    int p_o = j & 3;
    // phm_rule flat (32,4) -> (8,4,4): [q][p_i][p_o] = flat[q*16 + p_i*4 + p_o]
    float r = rule[((n & 7) << 4) + (p_i << 2) + p_o];
    Bhat[idx] = r * Wl[(n << 8) + i];
}

// -------------------------------------------------------------------------
// Kernel 2: T(128 x 384) = X(128 x 1024) @ Bhat^T   via v_wmma_f32_16x16x4_f32
// One wave per 16x16 tile of T; K loop of 256 WMMAs.
//   A lane layout: a[v] = A[M = l%16][K = 2*(l/16) + v]
//   B lane layout: b[v] = B[K = 2*(l/16) + v][N = l%16]
//   D lane layout: d[v] = D[M = 8*(l/16) + v][N = l%16]
// -------------------------------------------------------------------------
__global__ void phm_gemm1(const float* __restrict__ X,
                          const float* __restrict__ Bhat,
                          float* __restrict__ T) {
    int wave  = (blockIdx.x * blockDim.x + threadIdx.x) >> 5;  // 0..191
    int lane  = threadIdx.x & 31;
    int tileN = wave % (JDIM / 16);   // 0..23
    int tileM = wave / (JDIM / 16);   // 0..7
    int lrow  = lane & 15;
    int lhalf = lane >> 4;

    // A: row (tileM*16 + lrow) of X; pairs (4k+2*lhalf, +1) -> v2f index 2k+lhalf
    const v2f* xv = (const v2f*)(X + (size_t)(tileM * 16 + lrow) * IN_F);
    // B: column j = tileN*16 + lrow of T == row j of Bhat (j-major storage)
    const v2f* bv = (const v2f*)(Bhat + (size_t)(tileN * 16 + lrow) * IN_F);

    v8f c = {0.f, 0.f, 0.f, 0.f, 0.f, 0.f, 0.f, 0.f};
#pragma unroll 4
    for (int k = 0; k < IN_F / 4; ++k) {
        v2f a = xv[2 * k + lhalf];
        v2f b = bv[2 * k + lhalf];
        c = __builtin_amdgcn_wmma_f32_16x16x4_f32(
                /*neg_a=*/false, a, /*neg_b=*/false, b,
                /*c_mod=*/(short)0, c, /*reuse_a=*/false, /*reuse_b=*/false);
    }

    int mbase = tileM * 16 + 8 * lhalf;
    float* trow = T + tileN * 16 + lrow;
#pragma unroll
    for (int v = 0; v < 8; ++v)
        trow[(size_t)(mbase + v) * JDIM] = c[v];
}

// -------------------------------------------------------------------------
// Kernel 3: per (m, p_o): Y_block(128 x 256) = T_block(128 x 4) @ Wr(4 x 256)
// K = 4 -> exactly ONE v_wmma_f32_16x16x4_f32 per 16x16 output tile.
// Adds bias and writes the final (128 x 24576) output.
// -------------------------------------------------------------------------
__global__ void phm_gemm2(const float* __restrict__ T,
                          const float* __restrict__ Wr,
                          const float* __restrict__ bias,
                          float* __restrict__ Y) {
    int wave  = (blockIdx.x * blockDim.x + threadIdx.x) >> 5;  // 0..12287
    int lane  = threadIdx.x & 31;
    int tileN =  wave        & 15;   // 16 tiles of 16 over OUT_P=256
    int tileM = (wave >> 4)  & 7;    // 8 tiles of 16 over 128 rows
    int p_o   = (wave >> 7)  & 3;
    int m     =  wave >> 9;          // 0..23
    int lrow  = lane & 15;
    int lhalf = lane >> 4;

    int n0 = m * 4 + 2 * lhalf;      // absolute block index for K = 2*lhalf
    // A: a[v] = T[row = tileM*16+lrow][j = (n0+v)*4 + p_o]
    const float* trow = T + (size_t)(tileM * 16 + lrow) * JDIM + p_o;
    v2f a;
    a.x = trow[(size_t)n0 * 4];
    a.y = trow[(size_t)(n0 + 1) * 4];
    // B: b[v] = Wr[n0+v][tileN*16 + lrow]   (W_right_para flat: (96,256))
    const float* wrc = Wr + tileN * 16 + lrow;
    v2f b;
    b.x = wrc[(size_t)n0 * OUT_P];
    b.y = wrc[(size_t)(n0 + 1) * OUT_P];

    v8f c = {0.f, 0.f, 0.f, 0.f, 0.f, 0.f, 0.f, 0.f};
    c = __builtin_amdgcn_wmma_f32_16x16x4_f32(
            false, a, false, b, (short)0, c, false, false);

    int col  = m * OUT_F + p_o * OUT_P + tileN * 16 + lrow;
    float bb = bias[col];
    int mbase = tileM * 16 + 8 * lhalf;
#pragma unroll
    for (int v = 0; v < 8; ++v)
        Y[(size_t)(mbase + v) * OUT_COLS + col] = c[v] + bb;
}

// -------------------------------------------------------------------------
extern "C" void kernel_launch(void* const* d_in, const int* in_sizes, int n_in,
                              void* d_out, int out_size, void* d_ws, size_t ws_size,
                              hipStream_t stream) {
    const float* x    = (const float*)d_in[0];  // (128, 1024)
    const float* Wl   = (const float*)d_in[1];  // (96*256, 1)
    const float* Wr   = (const float*)d_in[2];  // (96, 256)
    const float* rule = (const float*)d_in[3];  // (32, 4)
    const float* bias = (const float*)d_in[4];  // (24576,)
    float* out = (float*)d_out;                 // (128, 24576)

    // workspace: Bhat (384*1024 f32 = 1.5 MB), then T (128*384 f32 = 192 KB)
    float* Bhat = (float*)d_ws;
    float* T    = Bhat + (size_t)JDIM * IN_F;

    // 1) build folded weight Bhat (j-major)
    phm_build_B<<<(JDIM * IN_F) / 256, 256, 0, stream>>>(Wl, rule, Bhat);

    // 2) T = X @ Bhat^T : 192 tiles, 1 wave each, 8 waves/block
    phm_gemm1<<<(ROWS / 16) * (JDIM / 16) / 8, 256, 0, stream>>>(x, Bhat, T);

    // 3) Y = blockwise T @ Wr + bias : 12288 tiles, 8 waves/block
    phm_gemm2<<<12288 / 8, 256, 0, stream>>>(T, Wr, bias, out);
}